// HyperbolicAttHRel_69303592288545
// MI455X (gfx1250) — compile-verified
//
#include <hip/hip_runtime.h>
#include <hip/hip_bf16.h>
#include <math.h>

#define CURV  0.01f
#define SQRTC 0.1f
#define EPSF  1e-15f

typedef __attribute__((ext_vector_type(2))) float v2f;
typedef __attribute__((ext_vector_type(8))) float v8f;

__device__ __forceinline__ float wave_reduce_sum(float v) {
#pragma unroll
    for (int m = 16; m >= 1; m >>= 1)
        v += __shfl_xor(v, m, 32);
    return v;
}

__device__ __forceinline__ float artanh_clip(float x) {
    // x >= 0 in our usage; clip to 1 - 1e-7 like the reference
    x = fminf(x, 1.0f - 1e-7f);
    return 0.5f * logf((1.0f + x) / (1.0f - x));
}

// ---------------------------------------------------------------------------
// Kernel 1: rel_hyp = exp_map_zero(rel_embedding), zero-padded to Rpad rows.
// Also y2[j] = |rel_hyp_j|^2.  One wave per row.
// ---------------------------------------------------------------------------
__global__ __launch_bounds__(256) void relprep_kernel(
    const float* __restrict__ rel, float* __restrict__ relh,
    float* __restrict__ y2, int R, int Rpad) {
    const int wave = threadIdx.x >> 5;
    const int lane = threadIdx.x & 31;
    const int row  = blockIdx.x * 8 + wave;
    if (row >= Rpad) return;

    float e[8];
    if (row < R) {
        const float* rp = rel + (size_t)row * 256 + lane * 8;
        float4 a = *(const float4*)(rp);
        float4 b = *(const float4*)(rp + 4);
        e[0]=a.x; e[1]=a.y; e[2]=a.z; e[3]=a.w;
        e[4]=b.x; e[5]=b.y; e[6]=b.z; e[7]=b.w;
    } else {
#pragma unroll
        for (int t = 0; t < 8; ++t) e[t] = 0.0f;
    }
    float p = 0.0f;
#pragma unroll
    for (int t = 0; t < 8; ++t) p += e[t] * e[t];
    const float n2 = wave_reduce_sum(p);
    const float n  = fmaxf(sqrtf(n2), EPSF);
    const float f  = tanhf(SQRTC * n) / (SQRTC * n);

    float* op = relh + (size_t)row * 256 + lane * 8;
    float4 o0 = make_float4(f*e[0], f*e[1], f*e[2], f*e[3]);
    float4 o1 = make_float4(f*e[4], f*e[5], f*e[6], f*e[7]);
    *(float4*)(op)     = o0;
    *(float4*)(op + 4) = o1;
    if (lane == 0) y2[row] = f * f * n2;   // |rel_hyp|^2
}

// ---------------------------------------------------------------------------
// Kernel 2: full per-row pipeline -> q[b] (2048x256) and x2[b] = |q_b|^2.
// One wave per batch row; lane holds 8 contiguous elements.
// ---------------------------------------------------------------------------
__global__ __launch_bounds__(256) void batchprep_kernel(
    const float* __restrict__ ent, const int* __restrict__ trip,
    const float* __restrict__ grot, const float* __restrict__ gref,
    const float* __restrict__ attnw,
    float* __restrict__ q, float* __restrict__ x2out, int B) {
    const int wave = threadIdx.x >> 5;
    const int lane = threadIdx.x & 31;
    const int row  = blockIdx.x * 8 + wave;
    if (row >= B) return;

    const int si = trip[row * 3 + 0];
    const int oi = trip[row * 3 + 2];

    float se[8], oe[8];
    {
        const float* sp = ent + (size_t)si * 256 + lane * 8;
        const float* op = ent + (size_t)oi * 256 + lane * 8;
        float4 a = *(const float4*)(sp);     float4 b = *(const float4*)(sp + 4);
        se[0]=a.x; se[1]=a.y; se[2]=a.z; se[3]=a.w;
        se[4]=b.x; se[5]=b.y; se[6]=b.z; se[7]=b.w;
        float4 c = *(const float4*)(op);     float4 d = *(const float4*)(op + 4);
        oe[0]=c.x; oe[1]=c.y; oe[2]=c.z; oe[3]=c.w;
        oe[4]=d.x; oe[5]=d.y; oe[6]=d.z; oe[7]=d.w;
    }

    // log_map_zero for s and o
    float ps = 0.0f, po = 0.0f;
#pragma unroll
    for (int t = 0; t < 8; ++t) { ps += se[t]*se[t]; po += oe[t]*oe[t]; }
    const float ss2 = wave_reduce_sum(ps);          // |s_emb|^2
    const float oo2 = wave_reduce_sum(po);          // |o_emb|^2
    const float ns = fmaxf(sqrtf(ss2), EPSF);
    const float no = fmaxf(sqrtf(oo2), EPSF);
    const float fs = artanh_clip(SQRTC * ns) / (SQRTC * ns);
    const float fo = artanh_clip(SQRTC * no) / (SQRTC * no);
    float st[8], ot[8];
#pragma unroll
    for (int t = 0; t < 8; ++t) { st[t] = fs * se[t]; ot[t] = fo * oe[t]; }

    // attention: sigmoid([s_tan, o_tan] . w)
    float ad = 0.0f;
    {
        const float* ws = attnw + lane * 8;
        const float* wo = attnw + 256 + lane * 8;
        float4 a = *(const float4*)(ws);   float4 b = *(const float4*)(ws + 4);
        float4 c = *(const float4*)(wo);   float4 d = *(const float4*)(wo + 4);
        float w0[8] = {a.x,a.y,a.z,a.w,b.x,b.y,b.z,b.w};
        float w1[8] = {c.x,c.y,c.z,c.w,d.x,d.y,d.z,d.w};
#pragma unroll
        for (int t = 0; t < 8; ++t) ad += st[t]*w0[t] + ot[t]*w1[t];
    }
    ad = wave_reduce_sum(ad);
    const float att = 1.0f / (1.0f + expf(-ad));

    // Givens rotation / reflection on s_tan, attention mix
    float mixed[8];
    {
        float4 ar = *(const float4*)(grot + lane * 4);
        float4 af = *(const float4*)(gref + lane * 4);
        float angr[4] = {ar.x, ar.y, ar.z, ar.w};
        float angf[4] = {af.x, af.y, af.z, af.w};
#pragma unroll
        for (int t = 0; t < 4; ++t) {
            const float x1 = st[2*t], x2_ = st[2*t+1];
            const float cr = cosf(angr[t]), sr = sinf(angr[t]);
            const float cf = cosf(angf[t]), sf = sinf(angf[t]);
            const float rot1 = cr*x1 - sr*x2_, rot2 = sr*x1 + cr*x2_;
            const float rf1  = cf*x1 + sf*x2_, rf2  = sf*x1 - cf*x2_;
            mixed[2*t]   = att*rot1 + (1.0f-att)*rf1;
            mixed[2*t+1] = att*rot2 + (1.0f-att)*rf2;
        }
    }

    // exp_map_zero(mixed)
    float pm = 0.0f;
#pragma unroll
    for (int t = 0; t < 8; ++t) pm += mixed[t]*mixed[t];
    const float mm2 = wave_reduce_sum(pm);
    const float nm = fmaxf(sqrtf(mm2), EPSF);
    const float fm = tanhf(SQRTC * nm) / (SQRTC * nm);
    float mh[8];
#pragma unroll
    for (int t = 0; t < 8; ++t) mh[t] = fm * mixed[t];
    const float x2v = fm * fm * mm2;                // |mixed_hyp|^2

    // query = mobius_add(-mixed_hyp, o_emb)
    float pd = 0.0f;
#pragma unroll
    for (int t = 0; t < 8; ++t) pd += mh[t]*oe[t];
    const float dmo = wave_reduce_sum(pd);
    const float xy  = -dmo;
    const float y2v = oo2;
    const float alpha = 1.0f + 2.0f*CURV*xy + CURV*y2v;
    const float beta  = 1.0f - CURV*x2v;
    const float den   = fmaxf(1.0f + 2.0f*CURV*xy + CURV*CURV*x2v*y2v, EPSF);
    float qv[8];
    float pq = 0.0f;
#pragma unroll
    for (int t = 0; t < 8; ++t) {
        qv[t] = (alpha * (-mh[t]) + beta * oe[t]) / den;
        pq += qv[t]*qv[t];
    }
    const float q2 = wave_reduce_sum(pq);

    float* qp = q + (size_t)row * 256 + lane * 8;
    *(float4*)(qp)     = make_float4(qv[0], qv[1], qv[2], qv[3]);
    *(float4*)(qp + 4) = make_float4(qv[4], qv[5], qv[6], qv[7]);
    if (lane == 0) x2out[row] = q2;
}

// ---------------------------------------------------------------------------
// Kernel 3: pairwise scores via f32 WMMA.  One wave = one 16x16 output tile,
// K=256 via 64 x V_WMMA_F32_16X16X4_F32.  Epilogue applies the mobius
// distance formula from (dot, x2, y2) and adds rel_bias.
// ---------------------------------------------------------------------------
__global__ __launch_bounds__(256) void scores_wmma_kernel(
    const float* __restrict__ q, const float* __restrict__ x2,
    const float* __restrict__ relh, const float* __restrict__ y2,
    const float* __restrict__ bias, float* __restrict__ out,
    int R, int ntilesN) {
    const int wave = threadIdx.x >> 5;
    const int lane = threadIdx.x & 31;
    const int bn = blockIdx.x * 8 + wave;    // tile col index (over Rpad/16)
    const int bm = blockIdx.y;               // tile row index (over B/16)
    if (bn >= ntilesN) return;               // wave-uniform exit

    const int m = lane & 15;                 // A-row / B-col for this lane
    const int h = lane >> 4;                 // K half select (lanes 16-31: K+2)
    const float* ap = q    + ((size_t)(bm * 16 + m)) * 256 + 2 * h;
    const float* bp = relh + ((size_t)(bn * 16 + m)) * 256 + 2 * h;

    v8f acc = {0.0f, 0.0f, 0.0f, 0.0f, 0.0f, 0.0f, 0.0f, 0.0f};
#pragma unroll 8
    for (int k = 0; k < 256; k += 4) {
        v2f av = *(const v2f*)(ap + k);
        v2f bv = *(const v2f*)(bp + k);
        acc = __builtin_amdgcn_wmma_f32_16x16x4_f32(
            /*neg_a=*/false, av, /*neg_b=*/false, bv,
            /*c_mod=*/(short)0, acc, /*reuse_a=*/false, /*reuse_b=*/false);
    }

    const int j = bn * 16 + m;               // relation index (col)
    const bool valid = (j < R);
    const float y2j = y2[j];                 // padded region is 0 (safe)
    const float bj  = valid ? bias[j] : 0.0f;
    const int rowbase = bm * 16 + 8 * h;
#pragma unroll
    for (int r = 0; r < 8; ++r) {
        const int b = rowbase + r;           // batch row (C layout: M = r + 8h)
        const float dot = acc[r];            // q_b . rel_hyp_j
        const float x2b = x2[b];
        const float alpha = 1.0f - 2.0f*CURV*dot + CURV*y2j;
        const float beta  = 1.0f - CURV*x2b;
        const float den   = fmaxf(1.0f - 2.0f*CURV*dot + CURV*CURV*x2b*y2j, EPSF);
        const float numsq = alpha*alpha*x2b - 2.0f*alpha*beta*dot + beta*beta*y2j;
        const float d2    = numsq / (den * den);
        if (valid) out[(size_t)b * R + j] = bj - d2;
    }
}

// ---------------------------------------------------------------------------
extern "C" void kernel_launch(void* const* d_in, const int* in_sizes, int n_in,
                              void* d_out, int out_size, void* d_ws, size_t ws_size,
                              hipStream_t stream) {
    const float* ent   = (const float*)d_in[0];   // (20000, 256)
    const float* rel   = (const float*)d_in[1];   // (R, 256)
    const int*   trip  = (const int*)d_in[2];     // (B, 3) int32
    const float* grot  = (const float*)d_in[3];   // (128,)
    const float* gref  = (const float*)d_in[4];   // (128,)
    const float* attnw = (const float*)d_in[5];   // (512,)
    const float* bias  = (const float*)d_in[6];   // (R,)
    float* out = (float*)d_out;

    const int B = in_sizes[2] / 3;                // 2048
    const int R = in_sizes[6];                    // 500
    const int Rpad = (R + 15) & ~15;              // 512
    const int ntilesN = Rpad / 16;                // 32
    const int ntilesM = B / 16;                   // 128

    float* ws   = (float*)d_ws;
    float* relh = ws;                             // Rpad*256
    float* y2   = relh + (size_t)Rpad * 256;      // Rpad
    float* q    = y2 + Rpad;                      // B*256
    float* x2   = q + (size_t)B * 256;            // B

    relprep_kernel<<<(Rpad + 7) / 8, 256, 0, stream>>>(rel, relh, y2, R, Rpad);
    batchprep_kernel<<<(B + 7) / 8, 256, 0, stream>>>(ent, trip, grot, gref,
                                                      attnw, q, x2, B);
    dim3 g3((ntilesN + 7) / 8, ntilesM);
    scores_wmma_kernel<<<g3, 256, 0, stream>>>(q, x2, relh, y2, bias, out,
                                               R, ntilesN);
}